// baseline_model_49959059587668
// MI455X (gfx1250) — compile-verified
//
#include <hip/hip_runtime.h>
#include <hip/hip_bf16.h>

// Problem constants: B=8, N=2048, D=768, two inputs, fp32 everywhere.
#define BATCH 8
#define NSENT 2048
#define DDIM  768
#define ROWS_PER_INPUT (BATCH * NSENT)          // 16384
#define OUT_PER_BATCH  ((size_t)NSENT * NSENT)  // 4194304
#define OUT_PER_INPUT  ((size_t)BATCH * OUT_PER_BATCH)

typedef __attribute__((ext_vector_type(2))) float v2f;
typedef __attribute__((ext_vector_type(8))) float v8f;

// ---------------------------------------------------------------------------
// Stage 1: per-row dual dot product. One wave32 per row (both inputs fused).
//   ws layout: [inp][0][16384] = s_src, [inp][1][16384] = s_dst + bias
// ---------------------------------------------------------------------------
__global__ void rowdot_kernel(const float* __restrict__ H,
                              const float* __restrict__ Dh,
                              const float* __restrict__ W,
                              const float* __restrict__ bptr,
                              float* __restrict__ ws) {
  const int wave = (blockIdx.x * blockDim.x + threadIdx.x) >> 5;   // 0..32767
  const int lane = threadIdx.x & 31;
  const int inp  = wave >> 14;          // 0: H, 1: doc_sents_h
  const int row  = wave & (ROWS_PER_INPUT - 1);

  const float* __restrict__ xr = (inp ? Dh : H) + (size_t)row * DDIM;
  const float* __restrict__ wa = W;
  const float* __restrict__ wb = W + DDIM;

  float da = 0.0f, db = 0.0f;
#pragma unroll
  for (int c = 0; c < DDIM / (32 * 4); ++c) {      // 6 float4 chunks per lane
    const int k = (c * 32 + lane) * 4;
    const float4 x4 = *(const float4*)(xr + k);
    const float4 a4 = *(const float4*)(wa + k);
    const float4 b4 = *(const float4*)(wb + k);
    da += x4.x * a4.x + x4.y * a4.y + x4.z * a4.z + x4.w * a4.w;
    db += x4.x * b4.x + x4.y * b4.y + x4.z * b4.z + x4.w * b4.w;
  }
#pragma unroll
  for (int off = 16; off > 0; off >>= 1) {          // wave32 butterfly reduce
    da += __shfl_xor(da, off, 32);
    db += __shfl_xor(db, off, 32);
  }
  if (lane == 0) {
    float* s_src  = ws + inp * 2 * ROWS_PER_INPUT;
    float* s_dstb = s_src + ROWS_PER_INPUT;
    s_src[row]  = da;
    s_dstb[row] = db + bptr[0];
  }
}

// ---------------------------------------------------------------------------
// Stage 2: out[b,i,j] = relu(s_src[b,i] + s_dstb[b,j]) as a rank-2 WMMA:
//   A(16x4) row m = [s_src[m], 1, 0, 0], B(4x16) col n = [1, s_dstb[n], 0, 0]^T
//   => D = s_src[i] + s_dstb[j] for the whole 16x16 tile in one v_wmma.
// One wave handles a 16-row x 512-col strip (32 tiles). Loads are
// unconditional (addresses valid for all 32 lanes) with cndmask selects; per
// tile the body is 1 load + 1 wmma + relu into distinct temps + one 8-store
// clause. Low VGPR footprint keeps the allocator from recycling store temps.
// ---------------------------------------------------------------------------
__global__ void edge_tile_kernel(const float* __restrict__ ws,
                                 float* __restrict__ out) {
  const int wave = (blockIdx.x * blockDim.x + threadIdx.x) >> 5;   // 0..8191
  const int lane = threadIdx.x & 31;

  const int chunk   = wave & 3;           // 4 column chunks of 512
  const int rowtile = (wave >> 2) & 127;  // 128 row tiles of 16
  const int batch   = (wave >> 9) & 7;
  const int inp     = wave >> 12;         // 0..1

  const float* __restrict__ s_src  =
      ws + inp * 2 * ROWS_PER_INPUT + batch * NSENT;
  const float* __restrict__ s_dstb =
      ws + inp * 2 * ROWS_PER_INPUT + ROWS_PER_INPUT + batch * NSENT;

  const int row0    = rowtile * 16;
  const int colbase = chunk * 512;
  const bool lo     = lane < 16;
  const int  l16    = lane & 15;

  // A operand: lanes 0-15 carry K=0,1 ; lanes 16-31 carry K=2,3 (zeroed)
  const float asrc = s_src[row0 + l16];   // unconditional, all lanes valid
  v2f a;
  a.x = lo ? asrc : 0.0f;                 // K=0 : s_src[row]
  a.y = lo ? 1.0f : 0.0f;                 // K=1 : 1
  // B operand: lanes 0-15 carry K=0,1 for col N=lane ; lanes 16-31 zeroed
  v2f b;
  b.x = lo ? 1.0f : 0.0f;                 // K=0 row : all ones

  float* __restrict__ p = out + inp * OUT_PER_INPUT + batch * OUT_PER_BATCH
                        + (size_t)(row0 + ((lane >> 4) << 3)) * NSENT
                        + colbase + l16;

#pragma unroll 4
  for (int t = 0; t < 32; ++t) {
    const float byv = s_dstb[colbase + t * 16 + l16];  // unconditional load
    b.y = lo ? byv : 0.0f;                // K=1 row : s_dst + bias (cndmask)
    v8f c = {};
    v8f d = __builtin_amdgcn_wmma_f32_16x16x4_f32(
        /*neg_a=*/false, a, /*neg_b=*/false, b,
        /*c_mod=*/(short)0, c, /*reuse_a=*/false, /*reuse_b=*/false);
    float r[8];
#pragma unroll
    for (int i = 0; i < 8; ++i)           // ReLU into distinct temps
      r[i] = fmaxf(d[i], 0.0f);
#pragma unroll
    for (int i = 0; i < 8; ++i)           // one s_clause of 8 b32 stores
      p[(size_t)i * NSENT + t * 16] = r[i];
  }
}

// ---------------------------------------------------------------------------
extern "C" void kernel_launch(void* const* d_in, const int* in_sizes, int n_in,
                              void* d_out, int out_size, void* d_ws, size_t ws_size,
                              hipStream_t stream) {
  const float* H  = (const float*)d_in[0];   // (8,2048,768)
  const float* Dh = (const float*)d_in[1];   // (8,2048,768)
  const float* W  = (const float*)d_in[2];   // (1,1536)
  const float* bb = (const float*)d_in[3];   // (1,)
  float* out = (float*)d_out;                // 2 x (8,2048,2048) concat
  float* ws  = (float*)d_ws;                 // 65536 floats used (256 KB)

  // Stage 1: 32768 waves (one per row, both inputs) -> 4096 blocks x 256 thr
  rowdot_kernel<<<4096, 256, 0, stream>>>(H, Dh, W, bb, ws);

  // Stage 2: 8192 waves (2 inputs x 8 batches x 128 rowtiles x 4 col chunks)
  edge_tile_kernel<<<1024, 256, 0, stream>>>(ws, out);
}